// Self_Attn1_3882650435657
// MI455X (gfx1250) — compile-verified
//
#include <hip/hip_runtime.h>
#include <hip/hip_bf16.h>

#ifndef __has_builtin
#define __has_builtin(x) 0
#endif

#if __has_builtin(__builtin_amdgcn_global_load_async_to_lds_b128)
#define ATTN_HAVE_ASYNC_LDS 1
#else
#define ATTN_HAVE_ASYNC_LDS 0
#endif

#define ATTN_B   4
#define ATTN_N   4096      // W*H = 64*64
#define THREADS  256
#define WAVES    8         // 256 / wave32

// Native clang vector types (required by nontemporal/async builtins).
typedef float v4f __attribute__((ext_vector_type(4)));
typedef int   v4i __attribute__((ext_vector_type(4)));
typedef __attribute__((address_space(1))) v4i* global_v4i_ptr;
typedef __attribute__((address_space(3))) v4i* lds_v4i_ptr;

__device__ __forceinline__ void wait_async_all() {
#if __has_builtin(__builtin_amdgcn_s_wait_asynccnt)
    __builtin_amdgcn_s_wait_asynccnt(0);
#elif ATTN_HAVE_ASYNC_LDS
    asm volatile("s_wait_asynccnt 0x0" ::: "memory");
#endif
}

// One block = 8 waves; each wave computes one attention row i:
//   s = x[b,i];  m = s>=0 ? s*max(x) : s*min(x)          (rank-1 softmax max)
//   pass 1: sum = sum_j exp(s*x_j - m), wsum = sum_j x_j*exp(s*x_j - m)
//   pass 2: attn[b,i,j] = exp(s*x_j - (m + ln(sum)))     (normalized, no div)
//   out[b,i] = gamma * (wsum/sum) + s
__global__ __launch_bounds__(THREADS) void rank1_softmax_attn(
    const float* __restrict__ x, const float* __restrict__ gamma,
    float* __restrict__ out, float* __restrict__ attn)
{
    __shared__ __align__(16) float sx[ATTN_N];     // 16 KB: x[b, :]
    __shared__ float smax[WAVES], smin[WAVES];

    const int tid  = threadIdx.x;
    const int lane = tid & 31;
    const int wave = tid >> 5;
    const int b    = blockIdx.x >> 9;                       // 512 blocks / batch
    const int i    = ((blockIdx.x & 511) * WAVES) + wave;   // row in [0, N)
    const float* __restrict__ xb = x + (size_t)b * ATTN_N;

    // ---- stage x[b,:] into LDS (CDNA5 async global->LDS path) ----
#if ATTN_HAVE_ASYNC_LDS
    #pragma unroll
    for (int r = 0; r < 4; ++r) {                 // 4 rounds x 256 thr x 16B = 16KB
        const int idx = (r * THREADS + tid) * 4;
        float* gsrc = const_cast<float*>(xb + idx);
        __builtin_amdgcn_global_load_async_to_lds_b128(
            (global_v4i_ptr)gsrc,
            (lds_v4i_ptr)(sx + idx),
            0, 0);
    }
    wait_async_all();
#else
    {
        const v4f* __restrict__ xb4 = reinterpret_cast<const v4f*>(xb);
        v4f* sx4w = reinterpret_cast<v4f*>(sx);
        #pragma unroll
        for (int r = 0; r < 4; ++r) {
            const int idx = r * THREADS + tid;
            sx4w[idx] = xb4[idx];
        }
    }
#endif
    __syncthreads();

    // ---- block-wide max/min of x[b,:] (rank-1 softmax max trick) ----
    float vmax = -__builtin_inff(), vmin = __builtin_inff();
    #pragma unroll
    for (int r = 0; r < ATTN_N / THREADS; ++r) {  // 16 elems / thread
        const float v = sx[r * THREADS + tid];
        vmax = fmaxf(vmax, v);
        vmin = fminf(vmin, v);
    }
    #pragma unroll
    for (int off = 16; off > 0; off >>= 1) {      // wave32 reduction
        vmax = fmaxf(vmax, __shfl_xor(vmax, off, 32));
        vmin = fminf(vmin, __shfl_xor(vmin, off, 32));
    }
    if (lane == 0) { smax[wave] = vmax; smin[wave] = vmin; }
    __syncthreads();
    float xmax = smax[0], xmin = smin[0];
    #pragma unroll
    for (int w = 1; w < WAVES; ++w) {
        xmax = fmaxf(xmax, smax[w]);
        xmin = fminf(xmin, smin[w]);
    }

    const float s = sx[i];
    const float m = (s >= 0.0f) ? s * xmax : s * xmin;
    const v4f* __restrict__ sx4 = reinterpret_cast<const v4f*>(sx);

    // ---- pass 1: row sum & value-weighted sum ----
    float sum = 0.0f, wsum = 0.0f;
    for (int j4 = lane; j4 < ATTN_N / 4; j4 += 32) {
        const v4f v = sx4[j4];
        const float e0 = __expf(fmaf(s, v.x, -m));
        const float e1 = __expf(fmaf(s, v.y, -m));
        const float e2 = __expf(fmaf(s, v.z, -m));
        const float e3 = __expf(fmaf(s, v.w, -m));
        sum  += (e0 + e1) + (e2 + e3);
        wsum += (v.x * e0 + v.y * e1) + (v.z * e2 + v.w * e3);
    }
    #pragma unroll
    for (int off = 16; off > 0; off >>= 1) {
        sum  += __shfl_xor(sum,  off, 32);
        wsum += __shfl_xor(wsum, off, 32);
    }

    // ---- pass 2: write normalized attention row, streaming NT b128 stores ----
    const float corr = m + __logf(sum);
    v4f* __restrict__ arow =
        reinterpret_cast<v4f*>(attn + ((size_t)b * ATTN_N + i) * ATTN_N);
    for (int j4 = lane; j4 < ATTN_N / 4; j4 += 32) {
        const v4f v = sx4[j4];
        v4f a;
        a.x = __expf(fmaf(s, v.x, -corr));
        a.y = __expf(fmaf(s, v.y, -corr));
        a.z = __expf(fmaf(s, v.z, -corr));
        a.w = __expf(fmaf(s, v.w, -corr));
        __builtin_nontemporal_store(a, arow + j4);
    }

    if (lane == 0) {
        out[(size_t)b * ATTN_N + i] = gamma[0] * (wsum / sum) + s;
    }
}

extern "C" void kernel_launch(void* const* d_in, const int* in_sizes, int n_in,
                              void* d_out, int out_size, void* d_ws, size_t ws_size,
                              hipStream_t stream) {
    (void)in_sizes; (void)n_in; (void)d_ws; (void)ws_size; (void)out_size;
    const float* x     = (const float*)d_in[0];   // (4, 64, 64) f32
    const float* gamma = (const float*)d_in[1];   // scalar f32
    float* out  = (float*)d_out;                           // 16384 floats
    float* attn = (float*)d_out + (size_t)ATTN_B * ATTN_N; // 4*4096*4096 floats

    dim3 grid(ATTN_B * ATTN_N / WAVES);   // 2048 blocks, one row per wave
    rank1_softmax_attn<<<grid, dim3(THREADS), 0, stream>>>(x, gamma, out, attn);
}